// NMSDeploy_65128884076565
// MI455X (gfx1250) — compile-verified
//
#include <hip/hip_runtime.h>
#include <stdint.h>

// ---------------- problem constants (match reference) ----------------
static constexpr int   N        = 2000000;   // boxes
static constexpr int   C        = 10;        // classes
static constexpr int   K        = 4096;      // NMS_PRE
static constexpr int   POST     = 500;       // NMS_POST
static constexpr float NMS_TH   = 0.5f;
static constexpr float SC_TH    = 0.1f;
static constexpr int   BINS     = 4096;      // histogram bins on s^10 (uniformized)
static constexpr int   CAP      = 8192;      // candidate capacity (>= K + boundary bin)
static constexpr int   KW       = K / 32;    // 128 mask words per row

// ---------------- workspace layout (bytes, 256B aligned) ----------------
static constexpr size_t OFF_HIST   = 0;                        // BINS u32      (16 KB)
static constexpr size_t OFF_CNT    = 16384;                    // 1 u32
static constexpr size_t OFF_SEL    = 16640;                    // 1 u32
static constexpr size_t OFF_CAND_S = 16896;                    // CAP f32       (32 KB)
static constexpr size_t OFF_CAND_I = 49664;                    // CAP u32       (32 KB)
static constexpr size_t OFF_TOP_S  = 82432;                    // K f32         (16 KB)
static constexpr size_t OFF_TOP_I  = 98816;                    // K u32         (16 KB)
static constexpr size_t OFF_TOPBOX = 115200;                   // K*4 f32       (64 KB)
static constexpr size_t OFF_KEEP   = 180736;                   // KW u32        (512 B)
static constexpr size_t OFF_MASK   = 181248;                   // K*KW u32      (2 MB)
static constexpr size_t OFF_SCORES = 2278400;                  // N f32         (8 MB)
// total ~10.28 MB

typedef uint32_t u32x4 __attribute__((ext_vector_type(4)));
typedef uint32_t u32x8 __attribute__((ext_vector_type(8)));

// s in (0,1); s^10 is ~Uniform(0,1) for max-of-10-uniforms, so bins are balanced.
__device__ __forceinline__ int score_bin(float s) {
  float s2 = s * s, s4 = s2 * s2, s8 = s4 * s4;
  float u = s8 * s2;
  int b = (int)(u * (float)BINS);
  return b < 0 ? 0 : (b >= BINS ? BINS - 1 : b);
}

// ---------------- 1: init counters ----------------
__global__ void nms_init_kernel(uint32_t* __restrict__ hist, uint32_t* __restrict__ cnt) {
  int t = blockIdx.x * blockDim.x + threadIdx.x;
  if (t == 0) *cnt = 0u;
  for (int b = t; b < BINS; b += blockDim.x * gridDim.x) hist[b] = 0u;
}

// ---------------- 2: max over classes + score spill + LDS histogram ----------------
__global__ __launch_bounds__(256) void nms_score_hist_kernel(const float* __restrict__ cls,
                                                             float* __restrict__ scores,
                                                             uint32_t* __restrict__ hist) {
  __shared__ uint32_t lhist[BINS];
  for (int b = threadIdx.x; b < BINS; b += blockDim.x) lhist[b] = 0u;
  __syncthreads();

  const int n4 = N / 4;
  for (int g = blockIdx.x * blockDim.x + threadIdx.x; g < n4; g += gridDim.x * blockDim.x) {
    float4 m = ((const float4*)cls)[g];
    #pragma unroll
    for (int c = 1; c < C; ++c) {
      float4 v = ((const float4*)(cls + (size_t)c * N))[g];
      m.x = fmaxf(m.x, v.x); m.y = fmaxf(m.y, v.y);
      m.z = fmaxf(m.z, v.z); m.w = fmaxf(m.w, v.w);
    }
    ((float4*)scores)[g] = m;
    if (m.x >= SC_TH) atomicAdd(&lhist[score_bin(m.x)], 1u);
    if (m.y >= SC_TH) atomicAdd(&lhist[score_bin(m.y)], 1u);
    if (m.z >= SC_TH) atomicAdd(&lhist[score_bin(m.z)], 1u);
    if (m.w >= SC_TH) atomicAdd(&lhist[score_bin(m.w)], 1u);
  }
  __syncthreads();
  for (int b = threadIdx.x; b < BINS; b += blockDim.x) {
    uint32_t v = lhist[b];
    if (v) atomicAdd(hist + b, v);
  }
}

// ---------------- 3: suffix scan -> threshold bin ----------------
__global__ void nms_select_kernel(const uint32_t* __restrict__ hist, uint32_t* __restrict__ sel) {
  if (threadIdx.x == 0) {
    uint32_t acc = 0; int selBin = 0;
    for (int b = BINS - 1; b >= 0; --b) {
      acc += hist[b];
      if (acc >= (uint32_t)K) { selBin = b; break; }
    }
    sel[0] = (uint32_t)selBin;
  }
}

// ---------------- 4: compact candidates (score >= thresh, bin >= selBin) ----------------
__global__ __launch_bounds__(256) void nms_compact_kernel(const float* __restrict__ scores,
                                                          const uint32_t* __restrict__ sel,
                                                          uint32_t* __restrict__ cnt,
                                                          float* __restrict__ cand_s,
                                                          uint32_t* __restrict__ cand_i) {
  const int selBin = (int)sel[0];
  for (int i = blockIdx.x * blockDim.x + threadIdx.x; i < N; i += gridDim.x * blockDim.x) {
    float s = scores[i];
    if (s >= SC_TH && score_bin(s) >= selBin) {
      uint32_t pos = atomicAdd(cnt, 1u);
      if (pos < (uint32_t)CAP) { cand_s[pos] = s; cand_i[pos] = (uint32_t)i; }
    }
  }
}

// ---------------- 5: one-block bitonic sort (desc score, asc idx) + box gather ----------------
__global__ __launch_bounds__(1024) void nms_sort_gather_kernel(const float* __restrict__ box,
                                                               const uint32_t* __restrict__ cnt_p,
                                                               const float* __restrict__ cand_s,
                                                               const uint32_t* __restrict__ cand_i,
                                                               float* __restrict__ top_s,
                                                               uint32_t* __restrict__ top_i,
                                                               float* __restrict__ top_box) {
  __shared__ float    s_key[CAP];   // 32 KB
  __shared__ uint32_t s_idx[CAP];   // 32 KB
  uint32_t cnt = *cnt_p; if (cnt > (uint32_t)CAP) cnt = CAP;

  for (int p = threadIdx.x; p < CAP; p += blockDim.x) {
    if (p < (int)cnt) { s_key[p] = cand_s[p]; s_idx[p] = cand_i[p]; }
    else              { s_key[p] = -1e30f;    s_idx[p] = 0xFFFFFFFFu; }
  }
  __syncthreads();

  for (int k = 2; k <= CAP; k <<= 1) {
    for (int j = k >> 1; j > 0; j >>= 1) {
      for (int i = threadIdx.x; i < CAP; i += blockDim.x) {
        int l = i ^ j;
        if (l > i) {
          float    ki = s_key[i], kl = s_key[l];
          uint32_t ii = s_idx[i], il = s_idx[l];
          bool up = ((i & k) == 0);  // this sub-block in final ("before") order
          bool lBeforeI = (kl > ki) || (kl == ki && il < ii);
          bool iBeforeL = (ki > kl) || (ki == kl && ii < il);
          if (up ? lBeforeI : iBeforeL) {
            s_key[i] = kl; s_key[l] = ki; s_idx[i] = il; s_idx[l] = ii;
          }
        }
      }
      __syncthreads();
    }
  }

  for (int p = threadIdx.x; p < K; p += blockDim.x) {
    bool real = (p < (int)cnt);
    float    s  = real ? s_key[p] : -1.0f;   // matches masked top_k padding value
    uint32_t id = real ? s_idx[p] : 0u;
    top_s[p] = s;
    top_i[p] = id;
    top_box[p * 4 + 0] = box[0 * (size_t)N + id];
    top_box[p * 4 + 1] = box[1 * (size_t)N + id];
    top_box[p * 4 + 2] = box[2 * (size_t)N + id];
    top_box[p * 4 + 3] = box[3 * (size_t)N + id];
  }
}

// ---------------- 6: IoU suppression bitmask; boxes staged to LDS via TDM ----------------
static constexpr int ROWS_PER_BLOCK = 32;

__global__ __launch_bounds__(1024) void nms_mask_kernel(const float* __restrict__ top_box_g,
                                                        uint32_t* __restrict__ mask) {
  __shared__ float4 lbox[K];  // 64 KB: all 4096 boxes, serves both row and column lookups

  // Wave 0 issues one TDM 1-D tile copy: 16384 f32 (64 KB) global -> LDS.
  if (threadIdx.x < 32) {
    uint64_t ga = (uint64_t)(uintptr_t)top_box_g;
    uint32_t la = (uint32_t)(uintptr_t)&lbox[0];          // LDS byte offset (addr[31:0])
    const uint32_t ELEMS = (uint32_t)(K * 4);             // 16384 elements of 4B
    // D# group 0: count=1 | lds_addr | global_addr[56:0] | type=2
    u32x4 g0 = { 1u, la, (uint32_t)ga,
                 (uint32_t)((ga >> 32) & 0x1FFFFFFull) | 0x80000000u };
    // D# group 1: data_size=4B, tensor_dim0=16384, tensor_dim1=1,
    //             tile_dim0=16384 (1-D tile), tensor_dim0_stride=16384
    u32x8 g1 = { 0x00020000u,                    // data_size=2 (4B) at [17:16]
                 (ELEMS & 0xFFFFu) << 16,        // tensor_dim0 low 16 -> bits[63:48]
                 (ELEMS >> 16) | (1u << 16),     // tensor_dim0 hi / tensor_dim1 low
                 (ELEMS << 16),                  // tensor_dim1 hi=0 | tile_dim0 at [127:112]
                 0u,                             // tile_dim1=0 (unused), tile_dim2=0
                 ELEMS,                          // tensor_dim0_stride low 32
                 0u, 0u };
    asm volatile("tensor_load_to_lds %0, %1" :: "s"(g0), "s"(g1) : "memory");
    __builtin_amdgcn_s_wait_tensorcnt(0);
  }
  __syncthreads();

  const int row0 = blockIdx.x * ROWS_PER_BLOCK;
  for (int t = threadIdx.x; t < ROWS_PER_BLOCK * KW; t += blockDim.x) {
    int lr = t >> 7;         // KW = 128 words/row
    int w  = t & (KW - 1);
    int i  = row0 + lr;
    float4 bi = lbox[i];
    float  ai = fmaxf(bi.z - bi.x, 0.f) * fmaxf(bi.w - bi.y, 0.f);
    uint32_t bits = 0u;
    int jbase = w << 5;
    #pragma unroll 8
    for (int b = 0; b < 32; ++b) {
      int j = jbase + b;
      float4 bj = lbox[j];
      float aj = fmaxf(bj.z - bj.x, 0.f) * fmaxf(bj.w - bj.y, 0.f);
      float iw = fmaxf(fminf(bi.z, bj.z) - fmaxf(bi.x, bj.x), 0.f);
      float ih = fmaxf(fminf(bi.w, bj.w) - fmaxf(bi.y, bj.y), 0.f);
      float inter = iw * ih;
      float iou = inter / (ai + aj - inter + 1e-8f);
      if ((j > i) && (iou > NMS_TH)) bits |= (1u << b);
    }
    mask[(size_t)i * KW + w] = bits;
  }
}

// ---------------- 7: serial suppression scan, single wave32 (barriers -> S_NOP) ----------------
__global__ __launch_bounds__(32) void nms_scan_kernel(const float* __restrict__ top_s,
                                                      const uint32_t* __restrict__ mask,
                                                      uint32_t* __restrict__ keep_words) {
  __shared__ uint32_t removed[KW];
  const int lane = threadIdx.x;  // 0..31, one wave
  for (int w = lane; w < KW; w += 32) {
    uint32_t valid = 0u;
    #pragma unroll 8
    for (int b = 0; b < 32; ++b)
      if (top_s[w * 32 + b] >= SC_TH) valid |= (1u << b);
    removed[w] = ~valid;   // start removed = ~keep0
  }
  __syncthreads();

  for (int i = 0; i < K; ++i) {
    uint32_t rw = removed[i >> 5];
    if (!((rw >> (i & 31)) & 1u)) {
      // row i survives: OR its suppression row into removed (4 words per lane, B128)
      uint4 m = ((const uint4*)(mask + (size_t)i * KW))[lane];
      removed[lane * 4 + 0] |= m.x;
      removed[lane * 4 + 1] |= m.y;
      removed[lane * 4 + 2] |= m.z;
      removed[lane * 4 + 3] |= m.w;
    }
    __syncthreads();  // single-wave workgroup: S_NOP per ISA, acts as compiler fence
  }

  for (int w = lane; w < KW; w += 32) keep_words[w] = ~removed[w];
}

// ---------------- 8: compact survivors, argmax labels, emit (500,6) ----------------
__global__ __launch_bounds__(512) void nms_final_kernel(const float* __restrict__ cls,
                                                        const uint32_t* __restrict__ keep_words,
                                                        const float* __restrict__ top_s,
                                                        const uint32_t* __restrict__ top_i,
                                                        const float* __restrict__ top_box,
                                                        float* __restrict__ out) {
  __shared__ int sel[POST];
  __shared__ int nsel;
  if (threadIdx.x == 0) {
    int n = 0;
    for (int k = 0; k < K && n < POST; ++k)
      if ((keep_words[k >> 5] >> (k & 31)) & 1u) sel[n++] = k;
    nsel = n;   // survivors are already score-descending: first 500 == top_k(POST)
  }
  __syncthreads();
  for (int r = threadIdx.x; r < POST; r += blockDim.x) {
    float o0 = 0.f, o1 = 0.f, o2 = 0.f, o3 = 0.f, o4 = 0.f, o5 = 0.f;
    if (r < nsel) {
      int k = sel[r];
      float s = top_s[k];
      if (s >= SC_TH) {
        o0 = top_box[k * 4 + 0]; o1 = top_box[k * 4 + 1];
        o2 = top_box[k * 4 + 2]; o3 = top_box[k * 4 + 3];
        o4 = s;
        uint32_t gid = top_i[k];
        float best = cls[gid]; int lab = 0;
        #pragma unroll
        for (int c = 1; c < C; ++c) {
          float v = cls[(size_t)c * N + gid];
          if (v > best) { best = v; lab = c; }  // first-max, matches jnp.argmax
        }
        o5 = (float)lab;
      }
    }
    out[r * 6 + 0] = o0; out[r * 6 + 1] = o1; out[r * 6 + 2] = o2;
    out[r * 6 + 3] = o3; out[r * 6 + 4] = o4; out[r * 6 + 5] = o5;
  }
}

// ---------------- host-side launcher ----------------
extern "C" void kernel_launch(void* const* d_in, const int* in_sizes, int n_in,
                              void* d_out, int out_size, void* d_ws, size_t ws_size,
                              hipStream_t stream) {
  (void)in_sizes; (void)n_in; (void)out_size; (void)ws_size;  // needs ~10.3 MB ws
  const float* box = (const float*)d_in[0];   // (1,4,N)
  const float* cls = (const float*)d_in[1];   // (1,C,N)
  float* out = (float*)d_out;                 // (1,POST,6)
  char* ws = (char*)d_ws;

  uint32_t* hist   = (uint32_t*)(ws + OFF_HIST);
  uint32_t* cnt    = (uint32_t*)(ws + OFF_CNT);
  uint32_t* sel    = (uint32_t*)(ws + OFF_SEL);
  float*    cand_s = (float*)   (ws + OFF_CAND_S);
  uint32_t* cand_i = (uint32_t*)(ws + OFF_CAND_I);
  float*    top_s  = (float*)   (ws + OFF_TOP_S);
  uint32_t* top_i  = (uint32_t*)(ws + OFF_TOP_I);
  float*    topbox = (float*)   (ws + OFF_TOPBOX);
  uint32_t* keep   = (uint32_t*)(ws + OFF_KEEP);
  uint32_t* mask   = (uint32_t*)(ws + OFF_MASK);
  float*    scores = (float*)   (ws + OFF_SCORES);

  nms_init_kernel<<<16, 256, 0, stream>>>(hist, cnt);
  nms_score_hist_kernel<<<2048, 256, 0, stream>>>(cls, scores, hist);
  nms_select_kernel<<<1, 32, 0, stream>>>(hist, sel);
  nms_compact_kernel<<<2048, 256, 0, stream>>>(scores, sel, cnt, cand_s, cand_i);
  nms_sort_gather_kernel<<<1, 1024, 0, stream>>>(box, cnt, cand_s, cand_i, top_s, top_i, topbox);
  nms_mask_kernel<<<K / ROWS_PER_BLOCK, 1024, 0, stream>>>(topbox, mask);
  nms_scan_kernel<<<1, 32, 0, stream>>>(top_s, mask, keep);
  nms_final_kernel<<<1, 512, 0, stream>>>(cls, keep, top_s, top_i, topbox, out);
}